// SlidingPosBiases3D_62560493633881
// MI455X (gfx1250) — compile-verified
//
#include <hip/hip_runtime.h>
#include <cmath>

#define BLOCK 256

typedef float v4f __attribute__((ext_vector_type(4)));

// ---------------------------------------------------------------------------
// Address-space cast helpers (generic -> LDS / global) for the async builtin.
// On AMDGPU the low 32 bits of a generic LDS pointer are the LDS byte offset,
// and generic global pointers are identical to AS(1) pointers.
// The builtin's parameters are int* in AS(1)/AS(3) (per hipcc diagnostic).
// ---------------------------------------------------------------------------
typedef __attribute__((address_space(3))) int lds_int_t;
typedef __attribute__((address_space(1))) int g1_int_t;

__device__ static inline lds_int_t* as_lds_i32(void* p) {
    return (lds_int_t*)(unsigned int)(unsigned long long)p;
}
__device__ static inline g1_int_t* as_g1_i32(const void* p) {
    return (g1_int_t*)(unsigned long long)p;
}

__device__ static inline void wait_async0() {
#if __has_builtin(__builtin_amdgcn_s_wait_asynccnt)
    __builtin_amdgcn_s_wait_asynccnt(0);
#else
    asm volatile("s_wait_asynccnt 0" ::: "memory");
#endif
}

// Stage n 32-bit words from global into LDS, one dword per lane per iteration.
__device__ static inline void stage32(void* dstL, const void* srcG, int n, int tid) {
#if __has_builtin(__builtin_amdgcn_global_load_async_to_lds_b32)
    const unsigned* src = (const unsigned*)srcG;
    unsigned* dst = (unsigned*)dstL;
    for (int i = tid; i < n; i += BLOCK) {
        __builtin_amdgcn_global_load_async_to_lds_b32(as_g1_i32(src + i), as_lds_i32(dst + i), 0, 0);
    }
#else
    const unsigned* src = (const unsigned*)srcG;
    unsigned* dst = (unsigned*)dstL;
    for (int i = tid; i < n; i += BLOCK) dst[i] = src[i];
#endif
}

// ---------------------------------------------------------------------------
// Kernel 1: precompute packed (h,w,d) decomposition for indices [0,N):
//   dec[i]  = h | (w << 10) | (d << 20)
// and (when D % 4 == 0) a compact stride-4 column table:
//   col4[c] = dec[4*c]   for c in [0, N/4)
// ---------------------------------------------------------------------------
__global__ void __launch_bounds__(256) dec_kernel(const int* __restrict__ Hp,
                                                  const int* __restrict__ Wp,
                                                  const int* __restrict__ Dp,
                                                  int* __restrict__ dec,
                                                  int* __restrict__ col4, int N) {
    (void)Hp;
    int i = blockIdx.x * blockDim.x + threadIdx.x;
    if (i >= N) return;
    int W = *Wp, D = *Dp;
    int d = i % D;
    int t = i / D;
    int w = t % W;
    int h = t / W;
    int packed = h | (w << 10) | (d << 20);
    dec[i] = packed;
    if (((D & 3) == 0) && ((i & 3) == 0)) col4[i >> 2] = packed;
}

// ---------------------------------------------------------------------------
// Kernel 2: one block per output row. Async-stage the bias volume into LDS
// (plus the full dec table only on the scalar fallback path), then emit the
// row with non-temporal float4 stores (pure store-bandwidth inner loop).
// ---------------------------------------------------------------------------
__global__ void __launch_bounds__(BLOCK) pos_bias_kernel(
    const float* __restrict__ biases, int bias_n, int B, int R,
    const int* __restrict__ Dp, const int* __restrict__ dec,
    const int* __restrict__ col4, float* __restrict__ out, int N) {

    extern __shared__ unsigned char smem[];
    float* biasL = (float*)smem;                          // bias_n floats
    int*   decL  = (int*)(smem + (size_t)bias_n * 4);     // N ints (fallback only)

    const int tid = threadIdx.x;
    const int D   = *Dp;
    const bool vec4 = ((D & 3) == 0);

    stage32(biasL, biases, bias_n, tid);
    if (!vec4) stage32(decL, dec, N, tid);
    wait_async0();
    __syncthreads();

    const int row   = blockIdx.x;
    const int rinfo = dec[row];          // uniform scalar load, L2-resident
    const int h1 = rinfo & 1023;
    const int w1 = (rinfo >> 10) & 1023;
    const int d1 = (rinfo >> 20) & 1023;
    const int S  = 2 * R;

    if (vec4) {
        // Vector path: D % 4 == 0 -> every aligned group of 4 columns shares
        // (h2, w2) and has consecutive d2 = d2b..d2b+3.
        const int N4 = N >> 2;
        v4f* out4 = (v4f*)(out + (size_t)row * (size_t)N);
        for (int c4 = tid; c4 < N4; c4 += BLOCK) {
            const int cinfo = col4[c4];  // L2-resident compact table
            const int h2  = cinfo & 1023;
            const int w2  = (cinfo >> 10) & 1023;
            const int d2b = (cinfo >> 20) & 1023;
            const int dh  = h2 - h1 + R;
            const int dw  = w2 - w1 + R;
            const int dd0 = d2b - d1 + R;
            v4f v = {0.f, 0.f, 0.f, 0.f};
            if ((unsigned)dh <= (unsigned)S && (unsigned)dw <= (unsigned)S) {
                const int base = (dh * B + dw) * B;
                if ((unsigned)(dd0)     <= (unsigned)S) v.x = biasL[base + dd0];
                if ((unsigned)(dd0 + 1) <= (unsigned)S) v.y = biasL[base + dd0 + 1];
                if ((unsigned)(dd0 + 2) <= (unsigned)S) v.z = biasL[base + dd0 + 2];
                if ((unsigned)(dd0 + 3) <= (unsigned)S) v.w = biasL[base + dd0 + 3];
            }
            __builtin_nontemporal_store(v, out4 + c4);
        }
    } else {
        // Scalar fallback for D not divisible by 4 (uses staged dec table).
        float* orow = out + (size_t)row * (size_t)N;
        for (int c = tid; c < N; c += BLOCK) {
            const int cinfo = decL[c];
            const int h2 = cinfo & 1023;
            const int w2 = (cinfo >> 10) & 1023;
            const int d2 = (cinfo >> 20) & 1023;
            const int dh = h2 - h1 + R;
            const int dw = w2 - w1 + R;
            const int dd = d2 - d1 + R;
            float v = 0.f;
            if ((unsigned)dh <= (unsigned)S && (unsigned)dw <= (unsigned)S &&
                (unsigned)dd <= (unsigned)S) {
                v = biasL[(dh * B + dw) * B + dd];
            }
            __builtin_nontemporal_store(v, orow + c);
        }
    }
}

extern "C" void kernel_launch(void* const* d_in, const int* in_sizes, int n_in,
                              void* d_out, int out_size, void* d_ws, size_t ws_size,
                              hipStream_t stream) {
    (void)n_in; (void)ws_size;
    const float* biases = (const float*)d_in[0];
    const int*   Hp     = (const int*)d_in[1];
    const int*   Wp     = (const int*)d_in[2];
    const int*   Dp     = (const int*)d_in[3];

    const int bias_n = in_sizes[0];                       // (2R+1)^3 = 3375
    const int B = (int)llround(cbrt((double)bias_n));     // 15
    const int R = (B - 1) / 2;                            // 7
    const int N = (int)llround(sqrt((double)out_size));   // 8000

    int* dec  = (int*)d_ws;       // N packed ints
    int* col4 = dec + N;          // N/4 packed ints (stride-4 subset)

    dec_kernel<<<(N + 255) / 256, 256, 0, stream>>>(Hp, Wp, Dp, dec, col4, N);

    // LDS: bias table + (worst case, scalar path) full dec table.
    const size_t shmem = (size_t)bias_n * 4 + (size_t)N * 4;  // ~45.5 KB
    pos_bias_kernel<<<N, BLOCK, shmem, stream>>>(biases, bias_n, B, R, Dp, dec,
                                                 col4, (float*)d_out, N);
}